// GraphConvCacheReuse_38543036514863
// MI455X (gfx1250) — compile-verified
//
#include <hip/hip_runtime.h>
#include <hip/hip_bf16.h>

// ---------------------------------------------------------------------------
// GraphConv + cache reuse for MI455X (gfx1250, wave32, WMMA).
// GEMM path uses V_WMMA_F32_16X16X4_F32 (fp32-exact, matches reference dtype).
// B operands staged in LDS K-pair-interleaved so each fragment is one
// ds_load_b64 into an even-aligned VGPR pair, conflict-free across half-waves.
// ---------------------------------------------------------------------------

typedef float v2f __attribute__((ext_vector_type(2)));
typedef float v8f __attribute__((ext_vector_type(8)));

#define IN_F   128
#define OUT_F  64
#define PSTR   160  // dwords per K-pair row: 160 % 64 == 32 -> half-waves hit disjoint banks

// ---------------- kernel 1: zero scratch (out_deg, in_deg, agg) -------------
__global__ void gc_zero_f4(float4* __restrict__ p, int n4) {
    int i = blockIdx.x * blockDim.x + threadIdx.x;
    int stride = gridDim.x * blockDim.x;
    float4 z; z.x = 0.f; z.y = 0.f; z.z = 0.f; z.w = 0.f;
    for (; i < n4; i += stride) p[i] = z;
}

// ---------------- kernel 2: degree histograms -------------------------------
__global__ void gc_degrees(const int* __restrict__ src, const int* __restrict__ dst,
                           float* __restrict__ out_deg, float* __restrict__ in_deg, int E) {
    int e = blockIdx.x * blockDim.x + threadIdx.x;
    if (e >= E) return;
    atomicAdd(&out_deg[src[e]], 1.0f);
    atomicAdd(&in_deg[dst[e]], 1.0f);
}

// ---------------- kernel 3: hw = diag(rsqrt(max(outdeg,1))) * feat * W ------
// One wave per 16-row tile. K loop: 128/4 = 32 WMMA steps; 4 N-tiles of 16.
__global__ __launch_bounds__(256) void gc_gemm_wmma(
        const float* __restrict__ feat, const float* __restrict__ W,
        const float* __restrict__ out_deg, float* __restrict__ hw, int nTiles) {
    // W [128x64] stored as K-pair interleaved rows:
    //   lw[(k>>1)*PSTR + 2*n + (k&1)] = W[k][n]
    __shared__ float lw[(IN_F / 2) * PSTR];
    for (int i = threadIdx.x; i < IN_F * OUT_F; i += 256) {
        int k = i >> 6, n = i & 63;
        lw[(k >> 1) * PSTR + (n << 1) + (k & 1)] = W[i];
    }
    __syncthreads();

    const int wave = threadIdx.x >> 5;
    const int lane = threadIdx.x & 31;
    const int tile = blockIdx.x * 8 + wave;
    if (tile >= nTiles) return;          // wave-uniform: EXEC stays all-1s for WMMA

    const int row0  = tile * 16;
    const int m     = lane & 15;         // A-matrix M per documented 16x4 f32 layout
    const int h     = lane >> 4;         // half-wave select
    const int khalf = h * 2;             // lanes 0-15: K pair {0,1}; lanes 16-31: {2,3}
    const int n     = lane & 15;         // B/D column

    const int   row   = row0 + m;
    const float scale = rsqrtf(fmaxf(out_deg[row], 1.0f));
    const float* __restrict__ arow = feat + (size_t)row * IN_F;

    v8f acc0 = {}; v8f acc1 = {}; v8f acc2 = {}; v8f acc3 = {};

    #pragma unroll 8
    for (int k0 = 0; k0 < IN_F; k0 += 4) {
        v2f a;
        a.x = arow[k0 + khalf]     * scale;
        a.y = arow[k0 + khalf + 1] * scale;

        // one ds_load_b64 per fragment: {W[k][n], W[k+1][n]} contiguous
        const v2f* __restrict__ bp = (const v2f*)(lw + ((k0 >> 1) + h) * PSTR);
        v2f b0 = bp[n];
        v2f b1 = bp[16 + n];
        v2f b2 = bp[32 + n];
        v2f b3 = bp[48 + n];

        acc0 = __builtin_amdgcn_wmma_f32_16x16x4_f32(false, a, false, b0, (short)0, acc0, false, false);
        acc1 = __builtin_amdgcn_wmma_f32_16x16x4_f32(false, a, false, b1, (short)0, acc1, false, false);
        acc2 = __builtin_amdgcn_wmma_f32_16x16x4_f32(false, a, false, b2, (short)0, acc2, false, false);
        acc3 = __builtin_amdgcn_wmma_f32_16x16x4_f32(false, a, false, b3, (short)0, acc3, false, false);
    }

    // C/D layout: lanes 0-15 -> M = vgpr (0..7); lanes 16-31 -> M = vgpr + 8; N = lane&15
    const int mbase = h * 8;
    float* __restrict__ obase = hw + (size_t)row0 * OUT_F;
    #pragma unroll
    for (int v = 0; v < 8; ++v) {
        float* orow = obase + (mbase + v) * OUT_F + n;
        orow[0]  = acc0[v];
        orow[16] = acc1[v];
        orow[32] = acc2[v];
        orow[48] = acc3[v];
    }
}

// ---------------- kernel 4: agg[dst] += hw[src] (scatter-add) ---------------
__global__ void gc_aggregate(const float* __restrict__ hw, const int* __restrict__ src,
                             const int* __restrict__ dst, float* __restrict__ agg, int E) {
    int idx = blockIdx.x * blockDim.x + threadIdx.x;
    if (idx >= E * 16) return;
    int e = idx >> 4, j = idx & 15;
    int s = src[e], d = dst[e];
    float4 v = ((const float4*)(hw + (size_t)s * OUT_F))[j];
    float* ap = agg + (size_t)d * OUT_F + j * 4;
    atomicAdd(ap + 0, v.x);
    atomicAdd(ap + 1, v.y);
    atomicAdd(ap + 2, v.z);
    atomicAdd(ap + 3, v.w);
}

__device__ __forceinline__ int gc_lower_bound(const int* __restrict__ a, int nel, int key) {
    int lo = 0, hi = nel;
    while (lo < hi) {
        int mid = (lo + hi) >> 1;
        if (a[mid] < key) lo = mid + 1; else hi = mid;
    }
    return lo;
}

// ---------------- kernel 5: normalize + bias + reuse scatter into full ------
__global__ void gc_finalize(const float* __restrict__ agg, const float* __restrict__ in_deg,
                            const float* __restrict__ bias, const float* __restrict__ reuse_emb,
                            const int* __restrict__ reuse_idx, int R,
                            float* __restrict__ out_full, int fullN) {
    int idx = blockIdx.x * blockDim.x + threadIdx.x;
    if (idx >= fullN * 16) return;
    int f = idx >> 4, j = idx & 15;
    int lb = gc_lower_bound(reuse_idx, R, f);
    float4* dst4 = (float4*)(out_full + (size_t)f * OUT_F) + j;
    if (lb < R && reuse_idx[lb] == f) {
        *dst4 = ((const float4*)(reuse_emb + (size_t)lb * OUT_F))[j];
    } else {
        int i = f - lb;   // node index: f minus count of pruned rows before f
        float s = rsqrtf(fmaxf(in_deg[i], 1.0f));
        float4 v = ((const float4*)(agg + (size_t)i * OUT_F))[j];
        float4 b = ((const float4*)bias)[j];
        float4 r;
        r.x = v.x * s + b.x; r.y = v.y * s + b.y;
        r.z = v.z * s + b.z; r.w = v.w * s + b.w;
        *dst4 = r;
    }
}

// ---------------- kernel 6: cache gather ------------------------------------
__global__ void gc_cache_gather(const float* __restrict__ out_full, const int* __restrict__ cache_idx,
                                float* __restrict__ out_cache, int C) {
    int idx = blockIdx.x * blockDim.x + threadIdx.x;
    if (idx >= C * 16) return;
    int c = idx >> 4, j = idx & 15;
    int fi = cache_idx[c];
    ((float4*)(out_cache + (size_t)c * OUT_F))[j] =
        ((const float4*)(out_full + (size_t)fi * OUT_F))[j];
}

// ---------------------------------------------------------------------------
extern "C" void kernel_launch(void* const* d_in, const int* in_sizes, int n_in,
                              void* d_out, int out_size, void* d_ws, size_t ws_size,
                              hipStream_t stream) {
    const float* feat      = (const float*)d_in[0];
    const float* weight    = (const float*)d_in[1];
    const float* bias      = (const float*)d_in[2];
    const float* reuse_emb = (const float*)d_in[3];
    const int*   src       = (const int*)d_in[4];
    const int*   dst       = (const int*)d_in[5];
    const int*   reuse_idx = (const int*)d_in[6];
    const int*   cache_idx = (const int*)d_in[7];

    const int N = in_sizes[0] / IN_F;     // 50000
    const int E = in_sizes[4];            // 800000
    const int R = in_sizes[6];            // 5000
    const int C = in_sizes[7];            // 10000
    const int fullN = N + R;              // 55000

    float* ws      = (float*)d_ws;
    float* out_deg = ws;                          // [N]
    float* in_deg  = ws + N;                      // [N]
    float* agg     = ws + 2 * (size_t)N;          // [N*64]
    float* hw      = ws + 2 * (size_t)N + 64 * (size_t)N;  // [N*64]

    float* out_full  = (float*)d_out;             // [fullN*64]
    float* out_cache = out_full + (size_t)fullN * OUT_F;   // [C*64]

    // 1) zero out_deg, in_deg, agg  (66*N floats, divisible by 4)
    {
        int n4 = (66 * N) / 4;
        int blocks = (n4 + 255) / 256;
        if (blocks > 4096) blocks = 4096;
        gc_zero_f4<<<blocks, 256, 0, stream>>>((float4*)ws, n4);
    }
    // 2) degrees
    gc_degrees<<<(E + 255) / 256, 256, 0, stream>>>(src, dst, out_deg, in_deg, E);
    // 3) WMMA GEMM with left normalization
    {
        int nTiles = (N + 15) / 16;               // 3125
        int blocks = (nTiles + 7) / 8;            // 8 waves/block
        gc_gemm_wmma<<<blocks, 256, 0, stream>>>(feat, weight, out_deg, hw, nTiles);
    }
    // 4) edge scatter-add
    {
        long long t = (long long)E * 16;
        gc_aggregate<<<(int)((t + 255) / 256), 256, 0, stream>>>(hw, src, dst, agg, E);
    }
    // 5) finalize into full
    {
        long long t = (long long)fullN * 16;
        gc_finalize<<<(int)((t + 255) / 256), 256, 0, stream>>>(agg, in_deg, bias, reuse_emb,
                                                                reuse_idx, R, out_full, fullN);
    }
    // 6) cache gather
    {
        long long t = (long long)C * 16;
        gc_cache_gather<<<(int)((t + 255) / 256), 256, 0, stream>>>(out_full, cache_idx, out_cache, C);
    }
}